// UTSContrastiveModel_29454885716559
// MI455X (gfx1250) — compile-verified
//
#include <hip/hip_runtime.h>
#include <hip/hip_bf16.h>

typedef __attribute__((ext_vector_type(2))) float        v2f;
typedef __attribute__((ext_vector_type(8))) float        v8f;
typedef __attribute__((ext_vector_type(4))) unsigned int u32x4;
typedef __attribute__((ext_vector_type(8))) int          i32x8;
typedef __attribute__((ext_vector_type(4))) int          i32x4;

#define DFEAT   128   // in_dim == hidden_dim
#define GBATCH  512   // graphs
#define PDIM    64    // proj dim
#define LAYERS  4

// ---------------------------------------------------------------- utilities

__global__ void copy_f4_kernel(float* __restrict__ dst, const float* __restrict__ src, int n4) {
  int i = blockIdx.x * blockDim.x + threadIdx.x;
  if (i < n4) ((float4*)dst)[i] = ((const float4*)src)[i];
}

__global__ void zero_f_kernel(float* __restrict__ p, int n) {
  int i = blockIdx.x * blockDim.x + threadIdx.x;
  if (i < n) p[i] = 0.0f;
}

// Rewrite W[K,OUT] row-major into WMMA-pair-major layout:
//   dst[(k/2)*2*OUT + j*2 + (k&1)] = src[k*OUT + j]
// so a B fragment (W[k][j], W[k+1][j]) is one aligned 8-byte LDS load.
__global__ void interleave_pairs_kernel(float* __restrict__ dst, const float* __restrict__ src,
                                        int K, int OUTc) {
  int i = blockIdx.x * blockDim.x + threadIdx.x;
  if (i >= K * OUTc) return;
  int k = i / OUTc, j = i - k * OUTc;
  dst[(k >> 1) * (2 * OUTc) + j * 2 + (k & 1)] = src[i];
}

// ------------------------------------------------------- edge scatter (GIN agg)
// One wave per edge: 32 lanes x float4 = 128 features. src/dst wave-uniform ->
// scalar loads; row gather = 4 coalesced b128 loads; accumulation buffer is
// 51.2MB (fits 192MB L2) so f32 atomics stay L2-resident.
__global__ void scatter_add_kernel(float* __restrict__ m, const float* __restrict__ h,
                                   const int* __restrict__ src, const int* __restrict__ dst,
                                   int E) {
  int t = blockIdx.x * blockDim.x + threadIdx.x;
  int e = t >> 5;
  if (e >= E) return;
  int c = (t & 31) * 4;
  int s = src[e];
  int d = dst[e];
  const float4 v = *(const float4*)(h + (size_t)s * DFEAT + c);
  float* p = m + (size_t)d * DFEAT + c;
  atomicAdd(p + 0, v.x);
  atomicAdd(p + 1, v.y);
  atomicAdd(p + 2, v.z);
  atomicAdd(p + 3, v.w);
}

// ------------------------------------------------------------- WMMA GEMM
// Out[nrows, OUT] = act(A[nrows, 128] @ W[128, OUT] + bias), fp32 WMMA.
// Block = 128 threads = 4 waves; each wave computes 16 rows x OUT cols.
// Weight tile (pair-interleaved, 128*OUT f32 <= 64KB) is DMA'd into LDS by the
// Tensor Data Mover (one tensor_load_to_lds + s_wait_tensorcnt) issued by wave 0.
//
// V_WMMA_F32_16X16X4_F32 fragment layouts (ISA 7.12.2):
//   A 16x4 : lane 0-15 -> M=lane, VGPR pair = K=(2*hi), K+1
//   B 4x16 : pair (K,K+1) per half-wave -> contiguous in pair-major LDS
//   C/D    : VGPR r: lanes0-15 M=r, lanes16-31 M=8+r, N = lane&15
template <int OUT, bool RELU>
__global__ __launch_bounds__(128) void gemm_wmma_kernel(const float* __restrict__ A,
                                                        const float* __restrict__ Wp,
                                                        const float* __restrict__ bias,
                                                        float* __restrict__ Out,
                                                        int nrows) {
  constexpr int NT = OUT / 16;  // 16-wide column tiles
  __shared__ float ldsW[DFEAT * OUT];
  __shared__ float ldsB[OUT];

  const int tid  = threadIdx.x;
  const int lane = tid & 31;
  const int wave = tid >> 5;

  if (tid < OUT) ldsB[tid] = bias[tid];

  // --- TDM: DMA the 1-D weight blob (128*OUT f32) global -> LDS -------------
  if (wave == 0) {
    constexpr unsigned elems = (unsigned)(DFEAT * OUT);  // f32 units, <= 16384
    unsigned long long ga = (unsigned long long)(uintptr_t)Wp;
    unsigned la = (unsigned)((unsigned long long)(uintptr_t)ldsW & 0xFFFFFFFFull);
    u32x4 g0 = {};
    g0[0] = 1u;                                              // count=1, user D#
    g0[1] = la;                                              // lds_addr
    g0[2] = (unsigned)(ga & 0xFFFFFFFFull);                  // global_addr[31:0]
    g0[3] = (unsigned)((ga >> 32) & 0x01FFFFFFull)           // global_addr[56:32]
            | (2u << 30);                                    // type = 2 ("image")
    i32x8 g1 = {};
    g1[0] = (int)(2u << 16);                                 // data_size = 4B
    g1[1] = (int)((elems & 0xFFFFu) << 16);                  // tensor_dim0 lo16
    g1[2] = (int)(((elems >> 16) & 0xFFFFu) | (1u << 16));   // dim0 hi | tensor_dim1=1
    g1[3] = (int)((elems & 0xFFFFu) << 16);                  // tile_dim0 = elems
    g1[4] = 1;                                               // tile_dim1 = 1
    g1[5] = (int)elems;                                      // tensor_dim0_stride
    i32x4 gz4 = {};                                          // groups 2/3 unused (<=2D)
    i32x8 gz8 = {};
    __builtin_amdgcn_tensor_load_to_lds(g0, g1, gz4, gz4, gz8, 0);
    __builtin_amdgcn_s_wait_tensorcnt(0);
  }
  __syncthreads();

  const int hi  = lane >> 4;   // half-wave select (K-pair / M+8)
  const int l15 = lane & 15;
  const int rowBase = (blockIdx.x * 4 + wave) * 16;

  // clamp A row for ragged tail (WMMA needs EXEC all-ones; stores are guarded)
  int rA = rowBase + l15;
  if (rA > nrows - 1) rA = nrows - 1;
  const float* Arow = A + (size_t)rA * DFEAT;

  v8f zero = {};
  v8f acc[NT];
#pragma unroll
  for (int t = 0; t < NT; ++t) acc[t] = zero;

#pragma unroll 4
  for (int k = 0; k < DFEAT; k += 4) {
    v2f a = *(const v2f*)(Arow + k + 2 * hi);               // global_load_b64
    __builtin_prefetch(Arow + k + 32, 0, 3);
    const float* bp = ldsW + (k / 2 + hi) * (2 * OUT) + l15 * 2;
#pragma unroll
    for (int t = 0; t < NT; ++t) {
      v2f b = *(const v2f*)(bp + t * 32);                   // ds_load_b64, no repack
      acc[t] = __builtin_amdgcn_wmma_f32_16x16x4_f32(
          /*neg_a=*/false, a, /*neg_b=*/false, b,
          /*c_mod=*/(short)0, acc[t], /*reuse_a=*/false, /*reuse_b=*/false);
    }
  }

  // epilogue: bias (+ReLU) and guarded store
#pragma unroll
  for (int t = 0; t < NT; ++t) {
    int col = t * 16 + l15;
    float bz = ldsB[col];
#pragma unroll
    for (int r = 0; r < 8; ++r) {
      int grow = rowBase + r + 8 * hi;
      if (grow < nrows) {
        float v = acc[t][r] + bz;
        if (RELU) v = fmaxf(v, 0.0f);
        Out[(size_t)grow * OUT + col] = v;
      }
    }
  }
}

// ------------------------------------------------------------- graph pooling

__global__ void count_nodes_kernel(float* __restrict__ cnt, const int* __restrict__ batch, int N) {
  int i = blockIdx.x * blockDim.x + threadIdx.x;
  if (i < N) atomicAdd(&cnt[batch[i]], 1.0f);
}

// one wave per node (same trick as scatter): batch[n] wave-uniform
__global__ void pool_sum_kernel(float* __restrict__ zsum, const float* __restrict__ h,
                                const int* __restrict__ batch, int N) {
  int t = blockIdx.x * blockDim.x + threadIdx.x;
  int n = t >> 5;
  if (n >= N) return;
  int c = (t & 31) * 4;
  int g = batch[n];
  const float4 v = *(const float4*)(h + (size_t)n * DFEAT + c);
  float* p = zsum + (size_t)g * DFEAT + c;
  atomicAdd(p + 0, v.x);
  atomicAdd(p + 1, v.y);
  atomicAdd(p + 2, v.z);
  atomicAdd(p + 3, v.w);
}

__global__ void pool_div_kernel(float* __restrict__ zg, const float* __restrict__ zsum,
                                const float* __restrict__ cnt, int n) {
  int i = blockIdx.x * blockDim.x + threadIdx.x;
  if (i < n) {
    float c = cnt[i / DFEAT];
    zg[i] = zsum[i] / fmaxf(c, 1.0f);
  }
}

// ---------------------------------------------------------------- launcher

extern "C" void kernel_launch(void* const* d_in, const int* in_sizes, int n_in,
                              void* d_out, int out_size, void* d_ws, size_t ws_size,
                              hipStream_t stream) {
  const float* x     = (const float*)d_in[0];   // [N,128]
  const int*   eidx  = (const int*)d_in[1];     // [2,E]: src row then dst row
  const int*   batch = (const int*)d_in[2];     // [N]
  const float* W1    = (const float*)d_in[3];   // [L,128,128]
  const float* b1    = (const float*)d_in[4];   // [L,128]
  const float* W2    = (const float*)d_in[5];   // [L,128,128]
  const float* b2    = (const float*)d_in[6];   // [L,128]
  const float* pW1   = (const float*)d_in[7];   // [128,128]
  const float* pb1   = (const float*)d_in[8];   // [128]
  const float* pW2   = (const float*)d_in[9];   // [128,64]
  const float* pb2   = (const float*)d_in[10];  // [64]

  const int N = in_sizes[0] / DFEAT;
  const int E = in_sizes[1] / 2;
  const int* srcIdx = eidx;
  const int* dstIdx = eidx + E;

  // output layout: h [N*128] | z_proj [G*64] | uts [L*N*128]
  float* out_h   = (float*)d_out;
  float* out_z   = out_h + (size_t)N * DFEAT;
  float* out_uts = out_z + (size_t)GBATCH * PDIM;

  // workspace layout
  const size_t WSZ = (size_t)DFEAT * DFEAT;            // 16384 floats per 128x128 W
  float* ws_m    = (float*)d_ws;                       // [N,128] aggregated message
  float* ws_t    = ws_m + (size_t)N * DFEAT;           // [N,128] mid activation
  float* ws_zsum = ws_t + (size_t)N * DFEAT;           // [G,128]
  float* ws_cnt  = ws_zsum + (size_t)GBATCH * DFEAT;   // [G]
  float* ws_zg   = ws_cnt + GBATCH;                    // [G,128] graph embedding
  float* ws_zh   = ws_zg + (size_t)GBATCH * DFEAT;     // [G,128] proj hidden
  float* ws_w1p  = ws_zh + (size_t)GBATCH * DFEAT;     // [L][128,128] interleaved W1
  float* ws_w2p  = ws_w1p + LAYERS * WSZ;              // [L][128,128] interleaved W2
  float* ws_pw1p = ws_w2p + LAYERS * WSZ;              // [128,128] interleaved pW1
  float* ws_pw2p = ws_pw1p + WSZ;                      // [128,64]  interleaved pW2

  const int thr = 256;
  const int copyBlocks    = (N * DFEAT / 4 + thr - 1) / thr;
  const int scatterBlocks = ((E * 32) + thr - 1) / thr;
  const int nodeBlocks32  = ((N * 32) + thr - 1) / thr;
  const int gemmBlocksN   = (N + 63) / 64;        // 4 waves x 16 rows per block
  const int gemmBlocksG   = (GBATCH + 63) / 64;
  const int ivBlocks128   = ((int)WSZ + thr - 1) / thr;
  const int ivBlocks64    = ((DFEAT * PDIM) + thr - 1) / thr;

  // one-time (per call) weight pair-interleave into workspace
  for (int l = 0; l < LAYERS; ++l) {
    interleave_pairs_kernel<<<ivBlocks128, thr, 0, stream>>>(
        ws_w1p + l * WSZ, W1 + (size_t)l * WSZ, DFEAT, DFEAT);
    interleave_pairs_kernel<<<ivBlocks128, thr, 0, stream>>>(
        ws_w2p + l * WSZ, W2 + (size_t)l * WSZ, DFEAT, DFEAT);
  }
  interleave_pairs_kernel<<<ivBlocks128, thr, 0, stream>>>(ws_pw1p, pW1, DFEAT, DFEAT);
  interleave_pairs_kernel<<<ivBlocks64, thr, 0, stream>>>(ws_pw2p, pW2, DFEAT, PDIM);

  const float* h_in = x;
  for (int l = 0; l < LAYERS; ++l) {
    float* h_out = out_uts + (size_t)l * N * DFEAT;
    // m = h_in
    copy_f4_kernel<<<copyBlocks, thr, 0, stream>>>(ws_m, h_in, N * DFEAT / 4);
    // m += segment_sum(h_in[src] -> dst)
    scatter_add_kernel<<<scatterBlocks, thr, 0, stream>>>(ws_m, h_in, srcIdx, dstIdx, E);
    // t = relu(m @ W1[l] + b1[l])
    gemm_wmma_kernel<DFEAT, true><<<gemmBlocksN, 128, 0, stream>>>(
        ws_m, ws_w1p + l * WSZ, b1 + (size_t)l * DFEAT, ws_t, N);
    // h = relu(t @ W2[l] + b2[l]) -> straight into uts[l]
    gemm_wmma_kernel<DFEAT, true><<<gemmBlocksN, 128, 0, stream>>>(
        ws_t, ws_w2p + l * WSZ, b2 + (size_t)l * DFEAT, h_out, N);
    h_in = h_out;
  }

  // final h output = uts[L-1]
  (void)hipMemcpyAsync(out_h, out_uts + (size_t)(LAYERS - 1) * N * DFEAT,
                       (size_t)N * DFEAT * sizeof(float), hipMemcpyDeviceToDevice, stream);

  // segment-mean pool over graphs
  zero_f_kernel<<<(GBATCH * DFEAT + thr - 1) / thr, thr, 0, stream>>>(ws_zsum, GBATCH * DFEAT);
  zero_f_kernel<<<(GBATCH + thr - 1) / thr, thr, 0, stream>>>(ws_cnt, GBATCH);
  count_nodes_kernel<<<(N + thr - 1) / thr, thr, 0, stream>>>(ws_cnt, batch, N);
  pool_sum_kernel<<<nodeBlocks32, thr, 0, stream>>>(ws_zsum, h_in, batch, N);
  pool_div_kernel<<<(GBATCH * DFEAT + thr - 1) / thr, thr, 0, stream>>>(
      ws_zg, ws_zsum, ws_cnt, GBATCH * DFEAT);

  // projection head: relu(zg @ pW1 + pb1) @ pW2 + pb2
  gemm_wmma_kernel<DFEAT, true><<<gemmBlocksG, 128, 0, stream>>>(ws_zg, ws_pw1p, pb1, ws_zh, GBATCH);
  gemm_wmma_kernel<PDIM, false><<<gemmBlocksG, 128, 0, stream>>>(ws_zh, ws_pw2p, pb2, out_z, GBATCH);
}